// BilateralBlur_12000138625578
// MI455X (gfx1250) — compile-verified
//
#include <hip/hip_runtime.h>
#include <hip/hip_bf16.h>

#define GAS __attribute__((address_space(1)))
#define LAS __attribute__((address_space(3)))

#ifndef __has_builtin
#define __has_builtin(x) 0
#endif

#if __has_builtin(__builtin_amdgcn_global_load_async_to_lds_b32)
#define USE_ASYNC_LDS 1
#else
#define USE_ASYNC_LDS 0
#endif

namespace {
constexpr int H = 256, W = 256, C = 4, KS = 13, R = 6;
constexpr int TILE = 16;
constexpr int HALO = TILE + KS - 1;   // 28
constexpr int HPIX = HALO * HALO;     // 784
// exp(-(dy^2+dx^2+d^2)/(2*3^2)) == exp2(K2 * (dy^2+dx^2+d^2))
constexpr float K2 = -0.080149737f;   // -0.5/(9*ln2)

__device__ __forceinline__ int refl(int i) {
  i = (i < 0) ? -i : i;               // numpy/jnp 'reflect': -1 -> 1
  return (i >= H) ? (2 * H - 2 - i) : i;  // H -> H-2
}

__device__ __forceinline__ float fast_exp2(float a) {
#if __has_builtin(__builtin_amdgcn_exp2f)
  return __builtin_amdgcn_exp2f(a);   // v_exp_f32
#else
  return exp2f(a);
#endif
}
}  // namespace

__global__ __launch_bounds__(256) void bilateral13_kernel(
    const float* __restrict__ x, float* __restrict__ out) {
  __shared__ float4 tile[HPIX];       // channel-interleaved halo tile

  const int tx = threadIdx.x, ty = threadIdx.y;
  const int tid = ty * TILE + tx;
  const int bx0 = blockIdx.x * TILE;
  const int by0 = blockIdx.y * TILE;
  const int b = blockIdx.z;
  const size_t plane = (size_t)H * W;
  float* ldsf = (float*)tile;

  // ---- Stage halo tile into LDS (async global->LDS on gfx1250) ----
  for (int s = tid; s < HPIX; s += TILE * TILE) {
    const int hy = s / HALO;
    const int hx = s - hy * HALO;
    const int gy = refl(by0 + hy - R);
    const int gx = refl(bx0 + hx - R);
    const size_t g0 = ((size_t)(b * C) * H + gy) * W + gx;
#if USE_ASYNC_LDS
#pragma unroll
    for (int c = 0; c < C; ++c) {
      __builtin_amdgcn_global_load_async_to_lds_b32(
          (GAS int*)(x + g0 + (size_t)c * plane),
          (LAS int*)(ldsf + s * 4 + c),
          /*offset=*/0, /*cpol=*/0);
    }
#else
#pragma unroll
    for (int c = 0; c < C; ++c) {
      ldsf[s * 4 + c] = x[g0 + (size_t)c * plane];
    }
#endif
  }

#if USE_ASYNC_LDS
#if __has_builtin(__builtin_amdgcn_s_wait_asynccnt)
  __builtin_amdgcn_s_wait_asynccnt(0);
#else
  asm volatile("s_wait_asynccnt 0" ::: "memory");
#endif
#endif
  __syncthreads();

  // ---- 13x13 bilateral accumulation from LDS ----
  const float4 cv = tile[(ty + R) * HALO + (tx + R)];
  float nx = 0.f, ny = 0.f, nz = 0.f, nw = 0.f, den = 0.f;

  for (int iy = 0; iy < KS; ++iy) {
    const int dy = iy - R;
    const float sy = (float)(dy * dy);
    const float4* row = &tile[(ty + iy) * HALO + tx];
#pragma unroll
    for (int ix = 0; ix < KS; ++ix) {
      const float4 p = row[ix];
      const int dx = ix - R;
      const float d = fabsf(p.x - cv.x) + fabsf(p.y - cv.y) +
                      fabsf(p.z - cv.z) + fabsf(p.w - cv.w);
      const float arg = K2 * (sy + (float)(dx * dx) + d * d);
      const float w = fast_exp2(arg);
      nx = fmaf(w, p.x, nx);
      ny = fmaf(w, p.y, ny);
      nz = fmaf(w, p.z, nz);
      nw = fmaf(w, p.w, nw);
      den += w;
    }
  }

  const float inv = 1.0f / den;       // den >= 1 (center tap weight == 1)
  const int gy = by0 + ty, gx = bx0 + tx;
  const size_t o0 = ((size_t)(b * C) * H + gy) * W + gx;
  out[o0 + 0 * plane] = nx * inv;
  out[o0 + 1 * plane] = ny * inv;
  out[o0 + 2 * plane] = nz * inv;
  out[o0 + 3 * plane] = nw * inv;
}

extern "C" void kernel_launch(void* const* d_in, const int* in_sizes, int n_in,
                              void* d_out, int out_size, void* d_ws, size_t ws_size,
                              hipStream_t stream) {
  (void)in_sizes; (void)n_in; (void)out_size; (void)d_ws; (void)ws_size;
  const float* x = (const float*)d_in[0];
  float* out = (float*)d_out;
  dim3 block(TILE, TILE, 1);          // 256 threads = 8 wave32
  dim3 grid(W / TILE, H / TILE, 2);   // 16 x 16 tiles x B=2
  bilateral13_kernel<<<grid, block, 0, stream>>>(x, out);
}